// ModelNew_36146444763147
// MI455X (gfx1250) — compile-verified
//
#include <hip/hip_runtime.h>
#include <cstdint>

#define T_STEPS 2048
#define BATCH   64
#define IN_DIM  256
#define HID     256
#define G3      768   // 3*H
#define LAYERS  3

typedef float v2f __attribute__((ext_vector_type(2)));
typedef float v8f __attribute__((ext_vector_type(8)));

// D(16x16 f32) = A(16x4 f32) * B(4x16 f32) + C   -- exact fp32 matrix op on CDNA5
static __device__ __forceinline__ v8f wmma_f32(v2f a, v2f b, v8f c) {
    return __builtin_amdgcn_wmma_f32_16x16x4_f32(false, a, false, b, (short)0, c, false, false);
}

static __device__ __forceinline__ v8f splat8(float s) {
    return (v8f){s, s, s, s, s, s, s, s};
}

// ---------------------------------------------------------------------------
// Input projection: Gx[M, 3H] = X[M, 256] @ W[3H, 256]^T + bias[3H]
// 2x2 register-blocked: each wave owns M-tiles {m0, m0+16} x N-tiles
// {n0, n0+16}; each (a0,a1,b0,b1) load quartet feeds 4 WMMAs.
// ---------------------------------------------------------------------------
__global__ __launch_bounds__(256) void gru_in_gemm(const float* __restrict__ X,
                                                   const float* __restrict__ W,
                                                   const float* __restrict__ bias,
                                                   float* __restrict__ Gx) {
    const int tid  = threadIdx.x;
    const int wave = tid >> 5;
    const int lane = tid & 31;
    const int r    = lane & 15;   // m (A) / n (B,D) within tile
    const int hh   = lane >> 4;   // K-half selector

    const int n0 = blockIdx.x * 32;                 // N-tile pair
    const int m0 = (blockIdx.y * 8 + wave) * 32;    // M-tile pair

    const float* ap0 = X + (size_t)(m0 + r) * IN_DIM + 2 * hh;  // a.x=A[m][k+2h]
    const float* ap1 = ap0 + (size_t)16 * IN_DIM;
    const float* bp0 = W + (size_t)(n0 + r) * IN_DIM + 2 * hh;  // b.x=B[k+2h][n]
    const float* bp1 = bp0 + (size_t)16 * IN_DIM;

    v8f c00 = splat8(bias[n0 + r]);
    v8f c01 = splat8(bias[n0 + 16 + r]);
    v8f c10 = c00;
    v8f c11 = c01;

    #pragma unroll 4
    for (int k = 0; k < IN_DIM; k += 4) {
        v2f a0 = *(const v2f*)(ap0 + k);
        v2f a1 = *(const v2f*)(ap1 + k);
        v2f b0 = *(const v2f*)(bp0 + k);
        v2f b1 = *(const v2f*)(bp1 + k);
        c00 = wmma_f32(a0, b0, c00);
        c01 = wmma_f32(a0, b1, c01);
        c10 = wmma_f32(a1, b0, c10);
        c11 = wmma_f32(a1, b1, c11);
    }

    // D: lane holds column n+r, rows m + 8*hh + v
    {
        float* d00 = Gx + (size_t)(m0 + 8 * hh) * G3 + n0 + r;
        float* d01 = d00 + 16;
        float* d10 = d00 + (size_t)16 * G3;
        float* d11 = d10 + 16;
        #pragma unroll
        for (int v = 0; v < 8; ++v) {
            d00[(size_t)v * G3] = c00[v];
            d01[(size_t)v * G3] = c01[v];
            d10[(size_t)v * G3] = c10[v];
            d11[(size_t)v * G3] = c11[v];
        }
    }
}

// ---------------------------------------------------------------------------
// Recurrent scan for one layer. Single workgroup (32 waves) keeps h in LDS.
// Per step: gh = h @ Whh^T + bhh via fp32 WMMA (A from LDS, reused over 6
// N-tiles per wave), gh staged to LDS, then fused gate update. Next step's
// gx slice is prefetched (global_prefetch_b8) while gate math runs.
// Dynamic LDS: h (64*256 f32 = 64KB) + gh (64*768 f32 = 192KB) = 256KB.
// ---------------------------------------------------------------------------
__global__ __launch_bounds__(1024) void gru_scan(const float* __restrict__ Gx,
                                                 const float* __restrict__ Whh,
                                                 const float* __restrict__ bhh,
                                                 const float* __restrict__ h0,
                                                 float* __restrict__ Out,
                                                 float* __restrict__ hT) {
    extern __shared__ float smem[];
    float* hL  = smem;                 // [BATCH, HID]
    float* ghL = smem + BATCH * HID;   // [BATCH, G3]

    const int tid  = threadIdx.x;
    const int wave = tid >> 5;
    const int lane = tid & 31;
    const int r    = lane & 15;
    const int hh   = lane >> 4;

    for (int e = tid; e < BATCH * HID; e += 1024) hL[e] = h0[e];
    __syncthreads();

    const int m0   = (wave & 3) * 16;   // 4 M-tiles cover B=64
    const int nset = wave >> 2;         // 8 groups x 6 N-tiles cover 3H=768

    float        bv[6];
    const float* bp[6];
    #pragma unroll
    for (int i = 0; i < 6; ++i) {
        const int n0 = (nset * 6 + i) * 16;
        bv[i] = bhh[n0 + r];
        bp[i] = Whh + (size_t)(n0 + r) * HID + 2 * hh;
    }
    const float* apL = hL + (m0 + r) * HID + 2 * hh;

    for (int t = 0; t < T_STEPS; ++t) {
        v8f c[6];
        #pragma unroll
        for (int i = 0; i < 6; ++i) c[i] = splat8(bv[i]);

        #pragma unroll 2
        for (int k = 0; k < HID; k += 4) {
            v2f a = *(const v2f*)(apL + k);        // ds_load_b64 from LDS
            #pragma unroll
            for (int i = 0; i < 6; ++i) {
                v2f b = *(const v2f*)(bp[i] + k);  // Whh stays hot in L2
                c[i] = wmma_f32(a, b, c[i]);
            }
        }

        // stage gh tiles into LDS for the cross-tile gate combine
        #pragma unroll
        for (int i = 0; i < 6; ++i) {
            const int n0 = (nset * 6 + i) * 16;
            float* gp = ghL + (m0 + 8 * hh) * G3 + n0 + r;
            #pragma unroll
            for (int v = 0; v < 8; ++v) gp[v * G3] = c[i][v];
        }
        __syncthreads();

        // prefetch next step's gx slice (192KB, one probe per 128B line)
        if (t + 1 < T_STEPS) {
            const float* gx_n = Gx + (size_t)(t + 1) * (BATCH * G3);
            for (int p = tid; p < (BATCH * G3) / 32; p += 1024)
                __builtin_prefetch(gx_n + (size_t)p * 32, 0, 0);
        }

        const float* gx_t  = Gx  + (size_t)t * (BATCH * G3);
        float*       out_t = Out + (size_t)t * (BATCH * HID);
        for (int e = tid; e < BATCH * HID; e += 1024) {
            const int b = e >> 8;
            const int j = e & 255;
            const size_t gxo = (size_t)b * G3 + j;
            const float xr = gx_t[gxo];
            const float xz = gx_t[gxo + HID];
            const float xn = gx_t[gxo + 2 * HID];
            const int go = b * G3 + j;
            const float hr = ghL[go];
            const float hz = ghL[go + HID];
            const float hn = ghL[go + 2 * HID];
            const float rg  = 1.0f / (1.0f + __expf(-(xr + hr)));
            const float zg  = 1.0f / (1.0f + __expf(-(xz + hz)));
            const float pre = xn + rg * hn;
            const float ng  = 2.0f / (1.0f + __expf(-2.0f * pre)) - 1.0f;  // tanh
            const float hp  = hL[e];
            const float hnew = (1.0f - zg) * ng + zg * hp;
            hL[e]    = hnew;   // disjoint per-thread element ownership
            out_t[e] = hnew;
        }
        __syncthreads();
    }

    for (int e = tid; e < BATCH * HID; e += 1024) hT[e] = hL[e];
}

// ---------------------------------------------------------------------------
// Host orchestration
// inputs: x[T,B,256], h0[L,B,256], W_ih[L,768,256], W_hh[L,768,256],
//         b_ih[L,768], b_hh[L,768]
// d_out : seq_out[T,B,256] ++ h_n[L,B,256]
// ws    : gx [M,768] (384MB) + ping buffer [M,256] (128MB)
// ---------------------------------------------------------------------------
extern "C" void kernel_launch(void* const* d_in, const int* in_sizes, int n_in,
                              void* d_out, int out_size, void* d_ws, size_t ws_size,
                              hipStream_t stream) {
    const float* x   = (const float*)d_in[0];
    const float* h0  = (const float*)d_in[1];
    const float* Wih = (const float*)d_in[2];
    const float* Whh = (const float*)d_in[3];
    const float* bih = (const float*)d_in[4];
    const float* bhh = (const float*)d_in[5];
    float* out = (float*)d_out;

    const size_t M = (size_t)T_STEPS * BATCH;   // 131072 rows
    float* gx   = (float*)d_ws;                 // M * G3 floats
    float* outA = gx + M * G3;                  // M * HID floats

    float* seq_out = out;                       // [T,B,H]
    float* hT_base = out + M * HID;             // [L,B,H]

    dim3 gemmGrid(G3 / 32, (unsigned)(M / 256));   // (24, 512)
    const size_t shmem = (size_t)(BATCH * HID + BATCH * G3) * sizeof(float);

    const float* layer_in = x;
    for (int l = 0; l < LAYERS; ++l) {
        gru_in_gemm<<<gemmGrid, 256, 0, stream>>>(
            layer_in, Wih + (size_t)l * G3 * IN_DIM, bih + (size_t)l * G3, gx);

        float* layer_out = (l == LAYERS - 1) ? seq_out : outA;
        gru_scan<<<1, 1024, shmem, stream>>>(
            gx, Whh + (size_t)l * G3 * HID, bhh + (size_t)l * G3,
            h0 + (size_t)l * BATCH * HID, layer_out,
            hT_base + (size_t)l * BATCH * HID);

        layer_in = layer_out;
    }
}